// GNNModelConstraints_20366734917651
// MI455X (gfx1250) — compile-verified
//
#include <hip/hip_runtime.h>

#define N_NODES 50000
#define N_EDGES 600000
#define HIDDEN  128
#define MTILES  (N_NODES / 16)   // 3125, exact

typedef float v2f __attribute__((ext_vector_type(2)));
typedef float v8f __attribute__((ext_vector_type(8)));

// ---------------- degree / normalization ----------------
__global__ void init_deg_kernel(float* deg) {
    int i = blockIdx.x * blockDim.x + threadIdx.x;
    if (i < N_NODES) deg[i] = 1.0f;               // self-loop weight
}

__global__ void scatter_deg_kernel(const int* dst, const float* ew, float* deg) {
    int e = blockIdx.x * blockDim.x + threadIdx.x;
    if (e < N_EDGES) atomicAdd(&deg[dst[e]], ew[e]);
}

__global__ void dinv_kernel(float* deg) {
    int i = blockIdx.x * blockDim.x + threadIdx.x;
    if (i < N_NODES) deg[i] = rsqrtf(deg[i]);     // in place: deg -> dinv
}

// ---------------- layer-1 feature transform ----------------
__global__ void xw1_kernel(const float* __restrict__ x, const float* __restrict__ W1,
                           float* __restrict__ xw) {
    int idx = blockIdx.x * blockDim.x + threadIdx.x;
    if (idx < N_NODES * HIDDEN) {
        int i = idx >> 7, h = idx & 127;
        xw[idx] = x[i] * W1[h];
    }
}

__global__ void zero_kernel(float* p, int n) {
    int i = blockIdx.x * blockDim.x + threadIdx.x;
    if (i < n) p[i] = 0.0f;
}

// ---------------- edge scatter: one wave per edge ----------------
__global__ void edge_scatter_kernel(const int* __restrict__ src, const int* __restrict__ dst,
                                    const float* __restrict__ ew, const float* __restrict__ dinv,
                                    const float* __restrict__ xw, float* __restrict__ agg) {
    int gid  = blockIdx.x * blockDim.x + threadIdx.x;
    int e    = gid >> 5;
    int lane = gid & 31;
    if (e >= N_EDGES) return;
    int s = src[e], d = dst[e];
    float norm = dinv[s] * ew[e] * dinv[d];
    const float4 v = ((const float4*)(xw + (size_t)s * HIDDEN))[lane];
    float* ag = agg + (size_t)d * HIDDEN + lane * 4;
    atomicAdd(ag + 0, v.x * norm);
    atomicAdd(ag + 1, v.y * norm);
    atomicAdd(ag + 2, v.z * norm);
    atomicAdd(ag + 3, v.w * norm);
}

// ---------------- fuse: agg + self-loop + bias, relu ----------------
__global__ void post_kernel(const float* __restrict__ agg, const float* __restrict__ xw,
                            const float* __restrict__ dinv, const float* __restrict__ b,
                            float* __restrict__ out) {
    int idx = blockIdx.x * blockDim.x + threadIdx.x;
    if (idx < N_NODES * HIDDEN) {
        int i = idx >> 7, h = idx & 127;
        float di = dinv[i];
        float v  = agg[idx] + xw[idx] * (di * di) + b[h];
        out[idx] = fmaxf(v, 0.0f);
    }
}

// ---------------- WMMA GEMM: D = A[N,128] x B[128,128], fp32 ----------------
// One wave computes a 16x128 output stripe: 8 accumulators of 16x16.
// K loop in steps of 4 using V_WMMA_F32_16X16X4_F32.
// A frag (16x4): lane<16 holds row m0+lane, K = k+0 (v.x), k+1 (v.y);
//                lane>=16 holds row m0+lane-16, K = k+2, k+3.
// B frag (4x16): lane picks column n0+(lane&15); same K-per-lane-half split.
// C/D: VGPR r holds M = m0 + r + (lane<16 ? 0 : 8), N = n0 + (lane&15).
__global__ void __launch_bounds__(128) gemm_wmma_kernel(const float* __restrict__ A,
                                                        const float* __restrict__ B,
                                                        float* __restrict__ D) {
    int wave = threadIdx.x >> 5;
    int lane = threadIdx.x & 31;
    int mtile = blockIdx.x * 4 + wave;
    if (mtile >= MTILES) return;                  // wave-uniform: EXEC stays all-ones
    int m0   = mtile * 16;
    int rowA = m0 + (lane & 15);
    int koff = (lane < 16) ? 0 : 2;
    int col  = lane & 15;

    v8f acc[8];
#pragma unroll
    for (int n = 0; n < 8; ++n) acc[n] = 0.0f;

#pragma unroll 4
    for (int k = 0; k < HIDDEN; k += 4) {
        v2f a = *(const v2f*)(A + (size_t)rowA * HIDDEN + k + koff);
        const float* brow0 = B + (size_t)(k + koff) * HIDDEN + col;
#pragma unroll
        for (int n = 0; n < 8; ++n) {
            v2f b;
            b.x = brow0[n * 16];
            b.y = brow0[HIDDEN + n * 16];
            acc[n] = __builtin_amdgcn_wmma_f32_16x16x4_f32(
                false, a, false, b, (short)0, acc[n], false, false);
        }
    }

    int rbase = m0 + ((lane < 16) ? 0 : 8);
#pragma unroll
    for (int n = 0; n < 8; ++n) {
        float* dp = D + (size_t)rbase * HIDDEN + n * 16 + col;
#pragma unroll
        for (int r = 0; r < 8; ++r) dp[(size_t)r * HIDDEN] = acc[n][r];
    }
}

// ---------------- final linear: one wave per node ----------------
__global__ void final_kernel(const float* __restrict__ h2, const float* __restrict__ Wl,
                             const float* __restrict__ bl, float* __restrict__ out) {
    int gid  = blockIdx.x * blockDim.x + threadIdx.x;
    int node = gid >> 5;
    int lane = gid & 31;
    if (node >= N_NODES) return;
    const float* hp = h2 + (size_t)node * HIDDEN;
    float s = 0.0f;
#pragma unroll
    for (int j = 0; j < 4; ++j) s += hp[lane + j * 32] * Wl[lane + j * 32];
#pragma unroll
    for (int off = 16; off > 0; off >>= 1) s += __shfl_xor(s, off, 32);
    if (lane == 0) out[node] = s + bl[0];
}

extern "C" void kernel_launch(void* const* d_in, const int* in_sizes, int n_in,
                              void* d_out, int out_size, void* d_ws, size_t ws_size,
                              hipStream_t stream) {
    const float* x   = (const float*)d_in[0];
    const int*   ei  = (const int*)d_in[1];
    const float* ew  = (const float*)d_in[2];
    const float* W1  = (const float*)d_in[3];
    const float* b1  = (const float*)d_in[4];
    const float* W2  = (const float*)d_in[5];
    const float* b2  = (const float*)d_in[6];
    const float* Wl  = (const float*)d_in[7];
    const float* bl  = (const float*)d_in[8];
    const int* src = ei;
    const int* dst = ei + N_EDGES;
    float* out = (float*)d_out;

    char* ws = (char*)d_ws;
    const size_t dinv_bytes = ((size_t)N_NODES * 4 + 255) & ~(size_t)255; // 200192
    const size_t nh = (size_t)N_NODES * HIDDEN;
    float* dinv = (float*)ws;
    float* bufA = (float*)(ws + dinv_bytes);       // xw1 / h1 / h2
    float* bufB = bufA + nh;                       // agg
    float* bufC = bufB + nh;                       // xw2

    const int T = 256;
    dim3 blk(T);
    int gN   = (N_NODES + T - 1) / T;
    int gE   = (N_EDGES + T - 1) / T;
    int gNH  = (int)((nh + T - 1) / T);
    int gE32 = (int)(((size_t)N_EDGES * 32 + T - 1) / T);
    int gN32 = (int)(((size_t)N_NODES * 32 + T - 1) / T);
    int gGemm = (MTILES + 3) / 4;

    // normalization
    init_deg_kernel<<<gN, blk, 0, stream>>>(dinv);
    scatter_deg_kernel<<<gE, blk, 0, stream>>>(dst, ew, dinv);
    dinv_kernel<<<gN, blk, 0, stream>>>(dinv);

    // layer 1
    xw1_kernel<<<gNH, blk, 0, stream>>>(x, W1, bufA);
    zero_kernel<<<gNH, blk, 0, stream>>>(bufB, (int)nh);
    edge_scatter_kernel<<<gE32, blk, 0, stream>>>(src, dst, ew, dinv, bufA, bufB);
    post_kernel<<<gNH, blk, 0, stream>>>(bufB, bufA, dinv, b1, bufA);   // h1 -> bufA

    // layer 2
    gemm_wmma_kernel<<<gGemm, dim3(128), 0, stream>>>(bufA, W2, bufC);  // xw2 -> bufC
    zero_kernel<<<gNH, blk, 0, stream>>>(bufB, (int)nh);
    edge_scatter_kernel<<<gE32, blk, 0, stream>>>(src, dst, ew, dinv, bufC, bufB);
    post_kernel<<<gNH, blk, 0, stream>>>(bufB, bufC, dinv, b2, bufA);   // h2 -> bufA

    // readout
    final_kernel<<<gN32, blk, 0, stream>>>(bufA, Wl, bl, out);
}